// LSTM_87771951661891
// MI455X (gfx1250) — compile-verified
//
#include <hip/hip_runtime.h>

typedef __attribute__((ext_vector_type(2))) float v2f;
typedef __attribute__((ext_vector_type(8))) float v8f;

#define B_   256
#define T_   512
#define H_   512
#define I_   256
#define NG_  2048            // 4*H
#define BH_  131072          // B*H
#define LDSS 516             // padded LDS row stride (floats): conflict-free for A b64 loads

__device__ __forceinline__ float sigf(float x) { return 1.0f / (1.0f + __expf(-x)); }

// ---------------- one-time: G0 = x0 @ W_w + W_b  (256 x 2048, K=256) ----------------
__global__ void g0_kernel(const float* __restrict__ x0, const float* __restrict__ Ww,
                          const float* __restrict__ Wb, float* __restrict__ G0) {
  int i = blockIdx.x * blockDim.x + threadIdx.x;
  if (i >= B_ * NG_) return;
  int b = i >> 11;
  int n = i & (NG_ - 1);
  float acc = Wb[n];
  const float* xr = x0 + b * I_;
  #pragma unroll 8
  for (int k = 0; k < I_; ++k) acc = fmaf(xr[k], Ww[k * NG_ + n], acc);
  G0[i] = acc;
}

// ---- one-time: permute U_w (H x 4H) into the exact per-lane WMMA-B operand order ----
// Uwp[nt][ks][lane] = float2{ Uw[k][n], Uw[k+1][n] },  k = 4*ks + 2*(lane>>4), n = 16*nt + (lane&15)
__global__ void uw_permute_kernel(const float* __restrict__ Uw, float* __restrict__ Uwp) {
  int i = blockIdx.x * blockDim.x + threadIdx.x;
  if (i >= 128 * 128 * 32) return;
  int lane = i & 31;
  int ks = (i >> 5) & 127;
  int nt = i >> 12;
  int k = ks * 4 + 2 * (lane >> 4);
  int n = nt * 16 + (lane & 15);
  float2 v;
  v.x = Uw[k * NG_ + n];
  v.y = Uw[(k + 1) * NG_ + n];
  reinterpret_cast<float2*>(Uwp)[i] = v;
}

// ---------------- per-timestep kernel ----------------
// grid: (32 H-chunks, 16 row-blocks), 128 threads = 4 waves (one gate tile i/f/g/o each).
// Each WG: stage 16 h-rows to LDS (osc applied to cols 0,1), 128x WMMA f32 16x16x4 per wave
// accumulating G0 + h@U_w, then the masked LSTM cell update for its 16x16 (b, j) patch.
__global__ __launch_bounds__(128) void lstm_step_kernel(
    const float* __restrict__ hcur, float* __restrict__ hnxt, float* __restrict__ cbuf,
    const float* __restrict__ G0, const float* __restrict__ Uwp,
    const int* __restrict__ bs_ptr, const float* __restrict__ tau_ptr,
    float* __restrict__ out, int t) {
  __shared__ float hA[16 * LDSS];
  __shared__ float gt[4][16][17];

  const int hc = blockIdx.x;          // H-chunk (16 cols)
  const int rb = blockIdx.y;          // row block (16 batch rows)
  const int tid = threadIdx.x;
  const int row0 = rb * 16;
  const int bs = bs_ptr[t];
  const float tau = tau_ptr[0];

  // stage h row-block -> LDS (coalesced b128), oscillator on cols 0,1 (t != 0)
  for (int i = tid; i < 16 * 128; i += 128) {
    int r = i >> 7, c4 = i & 127;
    float4 v = reinterpret_cast<const float4*>(hcur + (row0 + r) * H_)[c4];
    if (c4 == 0 && t != 0) {
      float a0 = v.x, a1 = v.y;
      v.x = a0 + tau * (1.5f * a0 + a1 * (1.0f / 1.5f));
      v.y = a1 + tau * (-1.5f * a0);
    }
    reinterpret_cast<float4*>(&hA[r * LDSS])[c4] = v;
  }
  __syncthreads();

  const int wave = tid >> 5, lane = tid & 31;
  const int lh = lane >> 4, ll = lane & 15;

  if (row0 < bs) {  // block-uniform: EXEC all-1s inside (WMMA requirement)
    const int ncol = wave * H_ + hc * 16 + ll;   // gate `wave`, this chunk's column
    v8f acc;
    #pragma unroll
    for (int r = 0; r < 8; ++r)                   // C init = G0 tile
      acc[r] = G0[(row0 + r + 8 * lh) * NG_ + ncol];

    const int nt = wave * 32 + hc;                // 16-wide N tile index (0..127)
    const float* bp = Uwp + ((size_t)nt * 4096 + lane) * 2;  // pre-permuted B stream
    const float* ap = &hA[ll * LDSS + 2 * lh];               // A: row=ll, K pair select
    #pragma unroll 4
    for (int ks = 0; ks < 128; ++ks) {
      v2f a = *reinterpret_cast<const v2f*>(ap + ks * 4);    // ds_load_b64, bank-clean
      v2f b = *reinterpret_cast<const v2f*>(bp + ks * 64);   // global_load_b64, coalesced
      acc = __builtin_amdgcn_wmma_f32_16x16x4_f32(false, a, false, b,
                                                  (short)0, acc, false, false);
    }
    #pragma unroll
    for (int r = 0; r < 8; ++r) gt[wave][r + 8 * lh][ll] = acc[r];
  }
  __syncthreads();

  // masked LSTM cell update for the 16x16 patch: 2 columns per thread (float2 I/O)
  {
    const int row = tid >> 3, col = (tid & 7) * 2;
    const int b = row0 + row;
    const int j = hc * 16 + col;
    const bool mask = b < bs;
    float iv0 = gt[0][row][col], iv1 = gt[0][row][col + 1];
    float fv0 = gt[1][row][col], fv1 = gt[1][row][col + 1];
    float gv0 = gt[2][row][col], gv1 = gt[2][row][col + 1];
    float ov0 = gt[3][row][col], ov1 = gt[3][row][col + 1];
    float2 cold = *reinterpret_cast<const float2*>(cbuf + b * H_ + j);
    float cn0 = sigf(fv0) * cold.x + sigf(iv0) * tanhf(gv0);
    float cn1 = sigf(fv1) * cold.y + sigf(iv1) * tanhf(gv1);
    float hn0 = sigf(ov0) * tanhf(cn0);
    float hn1 = sigf(ov1) * tanhf(cn1);
    float hold0 = hA[row * LDSS + j], hold1 = hA[row * LDSS + j + 1];
    float2 hw; hw.x = mask ? hn0 : hold0; hw.y = mask ? hn1 : hold1;
    *reinterpret_cast<float2*>(hnxt + b * H_ + j) = hw;
    float2 ow; ow.x = mask ? hn0 : 0.0f; ow.y = mask ? hn1 : 0.0f;
    *reinterpret_cast<float2*>(out + (size_t)t * BH_ + b * H_ + j) = ow;
    if (mask) {
      float2 cw; cw.x = cn0; cw.y = cn1;
      *reinterpret_cast<float2*>(cbuf + b * H_ + j) = cw;
    }
  }
}

extern "C" void kernel_launch(void* const* d_in, const int* in_sizes, int n_in,
                              void* d_out, int out_size, void* d_ws, size_t ws_size,
                              hipStream_t stream) {
  (void)in_sizes; (void)n_in; (void)out_size; (void)ws_size;
  const float* x_in  = (const float*)d_in[0];  // rnn_input_data (only first B rows used)
  const int*   bsz   = (const int*)  d_in[1];  // batch_sizes (T,)
  const float* h0    = (const float*)d_in[2];  // (1,B,H)
  const float* c0    = (const float*)d_in[3];  // (1,B,H)
  const float* tau   = (const float*)d_in[4];  // scalar
  const float* Ww    = (const float*)d_in[5];  // (I,4H)
  const float* Wb    = (const float*)d_in[6];  // (4H,)
  const float* Uw    = (const float*)d_in[7];  // (H,4H)
  float* out = (float*)d_out;

  // workspace layout (floats): G0 | Uwp | h ping | h pong | c   (~7.5 MB total)
  float* ws  = (float*)d_ws;
  float* G0  = ws;                  // 524288
  float* Uwp = G0 + B_ * NG_;       // 1048576
  float* hb0 = Uwp + 1048576;       // 131072
  float* hb1 = hb0 + BH_;           // 131072
  float* cb  = hb1 + BH_;           // 131072

  hipMemcpyAsync(hb0, h0, (size_t)BH_ * 4, hipMemcpyDeviceToDevice, stream);
  hipMemcpyAsync(cb,  c0, (size_t)BH_ * 4, hipMemcpyDeviceToDevice, stream);

  g0_kernel<<<(B_ * NG_ + 255) / 256, 256, 0, stream>>>(x_in, Ww, Wb, G0);
  uw_permute_kernel<<<(128 * 128 * 32 + 255) / 256, 256, 0, stream>>>(Uw, Uwp);

  float* hb[2] = {hb0, hb1};
  for (int t = 0; t < T_; ++t)
    lstm_step_kernel<<<dim3(32, 16), 128, 0, stream>>>(
        hb[t & 1], hb[(t + 1) & 1], cb, G0, Uwp, bsz, tau, out, t);

  // tail outputs: final h (back in hb0 after 512 steps) and c
  hipMemcpyAsync(out + (size_t)T_ * BH_,        hb0, (size_t)BH_ * 4, hipMemcpyDeviceToDevice, stream);
  hipMemcpyAsync(out + (size_t)T_ * BH_ + BH_,  cb,  (size_t)BH_ * 4, hipMemcpyDeviceToDevice, stream);
}